// BaseSeg_Balance_Prior_63831803953783
// MI455X (gfx1250) — compile-verified
//
#include <hip/hip_runtime.h>
#include <math.h>

// ---------------------------------------------------------------------------
// CDNA5 (gfx1250) fused point-MLP + prior EMA.
// fp32 WMMA (V_WMMA_F32_16X16X4_F32): exact-precision matrix path; kernel is
// output-BW bound (~83MB writes @ 23.3TB/s) so fp32 matrix throughput is ample.
// Each wave owns a 32x16 output strip (two 16x16 accumulators) so every
// B-fragment (weight) load feeds two WMMAs -> weights stream through L2 once
// per block.
// ---------------------------------------------------------------------------

typedef __attribute__((ext_vector_type(2))) float v2f;
typedef __attribute__((ext_vector_type(8))) float v8f;

#define B_      4
#define N_      40000
#define CIN     4
#define P_      (B_ * N_)        // 160000
#define EH      256              // enc hidden
#define ED      512              // enc out
#define PH      256              // proj hidden
#define PD      128              // prior dim
#define NC      13
#define MIN_PTS 256

#define TILE_M  32
#define THREADS 256
#define WAVES   8                // wave32: 256 threads = 8 waves

// LDS strides padded so (stride mod 64) == 4 dwords -> conflict-free frags
#define LD_A    260              // bufA: H1 / G  (32 x 256)
#define LD_B    516              // bufB: H2      (32 x 512), later F @ stride 132
#define LD_F    132
#define LD_X0   12               // X0 (32 x 8, cols 7 zero-padded)
#define LD_W1E  260              // staged W_enc1 (8 x 256, row 7 zero)

#define SM_BUFA 0
#define SM_BUFB (SM_BUFA + TILE_M * LD_A)            // 8320
#define SM_SCR  (SM_BUFB + TILE_M * LD_B)            // 24832
#define SM_X0   SM_SCR                                // 32*12  = 384
#define SM_W1E  (SM_SCR + TILE_M * LD_X0)            // 8*260  = 2080
#define SM_INVN (SM_SCR + TILE_M * LD_X0 + 8 * LD_W1E)
#define SM_TOT  (SM_INVN + TILE_M)                   // 27328 floats ~= 107 KB

static_assert(P_ % TILE_M == 0, "tile divides P");
static_assert(TILE_M == 32, "gemm_relu hardcodes a 2-high M-tile strip");

// Each wave computes one 32x16 strip per iteration: two 16x16 WMMA tiles
// (m0 = 0 and m0 = 16) sharing each B fragment.  K stepped by 4 with
// V_WMMA_F32_16X16X4_F32.  A comes from LDS; W (B operand) is row-major
// [K, N] (global, L2-resident weights, or LDS for layer 1).
__device__ __forceinline__ void gemm_relu(const float* A, int lda,
                                          const float* W, int ldw,
                                          const float* bias,
                                          float* D, int ldd,
                                          int Nt, int K,
                                          int lane, int wave)
{
    const int l15  = lane & 15;          // A row / B col / D col within tile
    const int koff = (lane >> 4) << 1;   // upper half-wave holds K+2,K+3
    const int rsel = (lane >> 4) << 3;   // D: upper half-wave holds rows M+8
    const float* arow0 = A + (size_t)l15 * lda + koff;         // rows 0..15
    const float* arow1 = A + (size_t)(16 + l15) * lda + koff;  // rows 16..31
    for (int nt = wave; nt < Nt; nt += WAVES) {
        const int n0 = nt << 4;
        const float* wcol = W + (size_t)koff * ldw + n0 + l15;
        v8f acc0 = {0.f, 0.f, 0.f, 0.f, 0.f, 0.f, 0.f, 0.f};
        v8f acc1 = {0.f, 0.f, 0.f, 0.f, 0.f, 0.f, 0.f, 0.f};
#pragma unroll 4
        for (int k0 = 0; k0 < K; k0 += 4) {
            v2f a0, a1, b;
            b.x  = wcol[(size_t)k0 * ldw];
            b.y  = wcol[(size_t)(k0 + 1) * ldw];
            a0.x = arow0[k0];
            a0.y = arow0[k0 + 1];
            a1.x = arow1[k0];
            a1.y = arow1[k0 + 1];
            acc0 = __builtin_amdgcn_wmma_f32_16x16x4_f32(
                       false, a0, false, b, (short)0, acc0, false, false);
            acc1 = __builtin_amdgcn_wmma_f32_16x16x4_f32(
                       false, a1, false, b, (short)0, acc1, false, false);
        }
        const float bc = bias[n0 + l15];
        float* drow0 = D + (size_t)rsel * ldd + n0 + l15;
        float* drow1 = D + (size_t)(16 + rsel) * ldd + n0 + l15;
#pragma unroll
        for (int v = 0; v < 8; ++v) {
            float d0 = acc0[v] + bc;
            float d1 = acc1[v] + bc;
            drow0[(size_t)v * ldd] = d0 > 0.f ? d0 : 0.f;   // fused ReLU
            drow1[(size_t)v * ldd] = d1 > 0.f ? d1 : 0.f;
        }
    }
}

// Pass 1: exact integer class histogram (deterministic).
__global__ __launch_bounds__(THREADS)
void hist_kernel(const int* __restrict__ y, int* __restrict__ gcount)
{
    __shared__ int h[NC];
    if (threadIdx.x < NC) h[threadIdx.x] = 0;
    __syncthreads();
    for (int i = blockIdx.x * blockDim.x + threadIdx.x; i < P_;
         i += gridDim.x * blockDim.x) {
        int l = y[i];
        if (l >= 0 && l < NC) atomicAdd(&h[l], 1);
    }
    __syncthreads();
    if (threadIdx.x < NC) atomicAdd(&gcount[threadIdx.x], h[threadIdx.x]);
}

// Pass 2: fully fused MLP chain + L2 normalize + current_prior + per-block
// deterministic class partial sums.
__global__ __launch_bounds__(THREADS)
void fused_mlp_kernel(const float* __restrict__ pos, const float* __restrict__ x,
                      const int* __restrict__ y,
                      const float* __restrict__ We1, const float* __restrict__ be1,
                      const float* __restrict__ We2, const float* __restrict__ be2,
                      const float* __restrict__ W1,  const float* __restrict__ b1,
                      const float* __restrict__ W2,  const float* __restrict__ b2,
                      const int* __restrict__ gcount,
                      float* __restrict__ partial, float* __restrict__ out)
{
    __shared__ float sm[SM_TOT];
    __shared__ int   slbl[TILE_M];
    const int tid  = threadIdx.x;
    const int lane = tid & 31;
    const int wave = tid >> 5;
    const int base = blockIdx.x * TILE_M;

    // zero X0 pad + W_enc1 pad rows
    for (int i = tid; i < TILE_M * LD_X0 + 8 * LD_W1E; i += THREADS)
        sm[SM_SCR + i] = 0.f;
    __syncthreads();

    // stage X0 = concat(features[4], pos[3], 0-pad)
    for (int i = tid; i < TILE_M * 7; i += THREADS) {
        int r = i / 7, c = i - r * 7;
        int p = base + r, b = p / N_, n = p - b * N_;
        float v = (c < CIN) ? x[((size_t)b * CIN + c) * N_ + n]
                            : pos[((size_t)b * N_ + n) * 3 + (c - CIN)];
        sm[SM_X0 + r * LD_X0 + c] = v;
    }
    // stage W_enc1 (7x256) into padded 8x260 (row 7 stays zero)
    for (int i = tid; i < 7 * EH; i += THREADS) {
        int r = i / EH, c = i - r * EH;
        sm[SM_W1E + r * LD_W1E + c] = We1[i];
    }
    if (tid < TILE_M) slbl[tid] = y[base + tid];
    __syncthreads();

    // L1: X0(32x8) @ We1(8x256)  -> bufA (H1)
    gemm_relu(&sm[SM_X0], LD_X0, &sm[SM_W1E], LD_W1E, be1,
              &sm[SM_BUFA], LD_A, EH / 16, 8, lane, wave);
    __syncthreads();
    // L2: H1 @ We2(256x512)      -> bufB (H2)
    gemm_relu(&sm[SM_BUFA], LD_A, We2, ED, be2,
              &sm[SM_BUFB], LD_B, ED / 16, EH, lane, wave);
    __syncthreads();
    // L3: H2 @ W1(512x256)       -> bufA (G)
    gemm_relu(&sm[SM_BUFB], LD_B, W1, PH, b1,
              &sm[SM_BUFA], LD_A, PH / 16, ED, lane, wave);
    __syncthreads();
    // L4: G @ W2(256x128)        -> bufB reused as F (stride 132)
    gemm_relu(&sm[SM_BUFA], LD_A, W2, PD, b2,
              &sm[SM_BUFB], LD_F, PD / 16, PH, lane, wave);
    __syncthreads();

    // per-row inverse L2 norm (matches v / max(||v||, 1e-12))
    if (tid < TILE_M) {
        const float* fr = &sm[SM_BUFB + tid * LD_F];
        float s = 0.f;
        for (int j = 0; j < PD; ++j) { float v = fr[j]; s += v * v; }
        sm[SM_INVN + tid] = 1.f / fmaxf(sqrtf(s), 1e-12f);
    }
    __syncthreads();

    // current_prior feature columns
    for (int i = tid; i < TILE_M * PD; i += THREADS) {
        int r = i >> 7, j = i & (PD - 1);
        float v = sm[SM_BUFB + r * LD_F + j] * sm[SM_INVN + r];
        out[(size_t)(base + r) * (PD + 1) + j] = v;
    }
    // label column with >=256 validity gate
    if (tid < TILE_M) {
        int l = slbl[tid];
        out[(size_t)(base + tid) * (PD + 1) + PD] =
            (gcount[l] >= MIN_PTS) ? (float)l : -1.f;
    }
    // deterministic per-block class partial sums (no float atomics)
    float* pb = partial + (size_t)blockIdx.x * (NC * PD);
    for (int i = tid; i < NC * PD; i += THREADS) {
        int c = i / PD, j = i - c * PD;
        float s = 0.f;
        for (int r = 0; r < TILE_M; ++r)
            if (slbl[r] == c)
                s += sm[SM_BUFB + r * LD_F + j] * sm[SM_INVN + r];
        pb[i] = s;
    }
}

// Pass 3: fixed-order partial reduction + EMA + L2 norm -> new_prior.
__global__ __launch_bounds__(PD)
void prior_kernel(const float* __restrict__ prior, const float* __restrict__ partial,
                  const int* __restrict__ gcount, float* __restrict__ out_np,
                  int nblocks)
{
    __shared__ float red[PD];
    const int r = blockIdx.x, j = threadIdx.x;
    float s = 0.f;
    for (int b = 0; b < nblocks; ++b)
        s += partial[(size_t)b * (NC * PD) + r * PD + j];
    const float cnt  = (float)gcount[r];
    const float mean = s / fmaxf(cnt, 1.f);
    const float pr   = prior[r * PD + j];
    const float cur  = (cnt >= (float)MIN_PTS) ? mean : pr;
    const float bl   = 0.999f * pr + (1.f - 0.999f) * cur;
    red[j] = bl * bl;
    __syncthreads();
    for (int st = PD / 2; st > 0; st >>= 1) {
        if (j < st) red[j] += red[j + st];
        __syncthreads();
    }
    out_np[r * PD + j] = bl / fmaxf(sqrtf(red[0]), 1e-12f);
}

extern "C" void kernel_launch(void* const* d_in, const int* in_sizes, int n_in,
                              void* d_out, int out_size, void* d_ws, size_t ws_size,
                              hipStream_t stream)
{
    const float* pos   = (const float*)d_in[0];
    const float* x     = (const float*)d_in[1];
    const int*   y     = (const int*)d_in[2];
    const float* We1   = (const float*)d_in[3];
    const float* be1   = (const float*)d_in[4];
    const float* We2   = (const float*)d_in[5];
    const float* be2   = (const float*)d_in[6];
    const float* W1    = (const float*)d_in[7];
    const float* b1    = (const float*)d_in[8];
    const float* W2    = (const float*)d_in[9];
    const float* b2    = (const float*)d_in[10];
    const float* prior = (const float*)d_in[11];
    float* out = (float*)d_out;

    int*   gcount  = (int*)d_ws;                          // 13 ints (zeroed)
    float* partial = (float*)((char*)d_ws + 256);         // nblocks * 13*128 f32

    const int nblocks = P_ / TILE_M;                      // 5000

    hipMemsetAsync(gcount, 0, 256, stream);
    hist_kernel<<<256, THREADS, 0, stream>>>(y, gcount);
    fused_mlp_kernel<<<nblocks, THREADS, 0, stream>>>(
        pos, x, y, We1, be1, We2, be2, W1, b1, W2, b2, gcount, partial, out);
    prior_kernel<<<NC, PD, 0, stream>>>(
        prior, partial, gcount, out + (size_t)P_ * (PD + 1), nblocks);
}